// Convolution_36833639530669
// MI455X (gfx1250) — compile-verified
//
#include <hip/hip_runtime.h>
#include <hip/hip_bf16.h>

// ---------------------------------------------------------------------------
// Sparse 2x2x2 generative conv, G=128, CIN=32, COUT=64.
// out blocks = occupied 64^3 half-res blocks, sorted by linear code
// (== np.unique lexicographic order). out_feat = sum_t feat[slot[t]] @ W[t].
// GEMM done with V_WMMA_F32_16X16X4_F32 (exact f32).
// ---------------------------------------------------------------------------

typedef float v2f __attribute__((ext_vector_type(2)));
typedef float v8f __attribute__((ext_vector_type(8)));

#define NBLK   (64 * 64 * 64)     // 262144 candidate output blocks
#define NGRP   256                // scan groups
#define PERGRP 1024               // blocks per group (NGRP*PERGRP == NBLK)

// ---- phase 0: init slot table to -1 ---------------------------------------
__global__ void k_init_slots(int* __restrict__ slots, int n) {
    int i = blockIdx.x * blockDim.x + threadIdx.x;
    if (i < n) slots[i] = -1;
}

// ---- phase 1: scatter input rows into block slots -------------------------
__global__ void k_scatter(const int* __restrict__ pos, int* __restrict__ slots, int n) {
    int i = blockIdx.x * blockDim.x + threadIdx.x;
    if (i >= n) return;
    int x = pos[i * 3 + 0], y = pos[i * 3 + 1], z = pos[i * 3 + 2];
    int bc   = ((x >> 1) << 12) | ((y >> 1) << 6) | (z >> 1);
    int slot = ((x & 1) << 2) | ((y & 1) << 1) | (z & 1);
    slots[bc * 8 + slot] = i;   // cells are unique -> no race
}

// ---- phase 2: per-group occupancy count -----------------------------------
__global__ __launch_bounds__(256) void k_count(const int* __restrict__ slots,
                                               unsigned char* __restrict__ occ,
                                               int* __restrict__ partials) {
    __shared__ int s[256];
    int g = blockIdx.x, t = threadIdx.x;
    int base = g * PERGRP + t * 4;
    int c = 0;
#pragma unroll
    for (int i = 0; i < 4; ++i) {
        int b = base + i;
        int o = 0;
#pragma unroll
        for (int s8 = 0; s8 < 8; ++s8) o |= (slots[b * 8 + s8] >= 0) ? 1 : 0;
        occ[b] = (unsigned char)o;
        c += o;
    }
    s[t] = c;
    __syncthreads();
    for (int off = 128; off > 0; off >>= 1) {
        if (t < off) s[t] += s[t + off];
        __syncthreads();
    }
    if (t == 0) partials[g] = s[0];
}

// ---- phase 3: order-preserving compaction + positions ---------------------
__global__ __launch_bounds__(256) void k_compact(const unsigned char* __restrict__ occ,
                                                 const int* __restrict__ partials,
                                                 int* __restrict__ blocks,
                                                 float* __restrict__ out_pos,
                                                 int M) {
    __shared__ int s[256];
    int g = blockIdx.x, t = threadIdx.x;

    // group offset = sum of partials of earlier groups
    s[t] = partials[t];
    __syncthreads();
    int goff = 0;
    for (int i = 0; i < g; ++i) goff += s[i];
    __syncthreads();

    int base = g * PERGRP + t * 4;
    int cl[4];
    int c = 0;
#pragma unroll
    for (int i = 0; i < 4; ++i) { cl[i] = occ[base + i]; c += cl[i]; }

    // Hillis-Steele inclusive scan over per-thread counts
    s[t] = c;
    __syncthreads();
    for (int off = 1; off < 256; off <<= 1) {
        int v = (t >= off) ? s[t - off] : 0;
        __syncthreads();
        s[t] += v;
        __syncthreads();
    }
    int idx = goff + s[t] - c;   // exclusive

#pragma unroll
    for (int i = 0; i < 4; ++i) {
        if (cl[i]) {
            int b = base + i;
            if (idx < M) {
                blocks[idx] = b;
                float bx = (float)(b >> 12);
                float by = (float)((b >> 6) & 63);
                float bz = (float)(b & 63);
                out_pos[idx * 3 + 0] = bx + 0.25f;   // (2b + 0.5) / 2
                out_pos[idx * 3 + 1] = by + 0.25f;
                out_pos[idx * 3 + 2] = bz + 0.25f;
            }
            ++idx;
        }
    }
}

// ---- phase 4: repack weight into B-fragment layout ------------------------
// bpack[((t*4+n)*8+q)*32 + lane] (float2):
//   { W[t][4q+2g][16n+col], W[t][4q+2g+1][16n+col] },  g=lane>>4, col=lane&15
__global__ void k_wpack(const float* __restrict__ w, float* __restrict__ bpack) {
    int tid = blockIdx.x * blockDim.x + threadIdx.x;
    if (tid >= 8 * 4 * 8 * 32) return;
    int lane = tid & 31;
    int q    = (tid >> 5) & 7;
    int n    = (tid >> 8) & 3;
    int t    = tid >> 10;
    int gg   = lane >> 4;
    int col  = lane & 15;
    int k0   = 4 * q + 2 * gg;
    int N    = 16 * n + col;
    bpack[tid * 2 + 0] = w[(t * 32 + k0 + 0) * 64 + N];
    bpack[tid * 2 + 1] = w[(t * 32 + k0 + 1) * 64 + N];
}

// ---- phase 5: WMMA GEMM ---------------------------------------------------
// One wave per 16-output tile; acc = 4 x (16x16 f32) covering 64 couts.
__global__ __launch_bounds__(256) void k_gemm(const float* __restrict__ feat,
                                              const int* __restrict__ slots,
                                              const int* __restrict__ blocks,
                                              const float* __restrict__ bpack,
                                              float* __restrict__ outf,
                                              int M) {
    int wave = threadIdx.x >> 5;
    int lane = threadIdx.x & 31;
    int tile = blockIdx.x * 8 + wave;
    int m0 = tile * 16;
    if (m0 >= M) return;               // wave-uniform exit: EXEC all-ones below

    int r = lane & 15;                 // output row within tile (A: M=row)
    int g = lane >> 4;                 // K-pair selector
    int m = m0 + r;
    bool rowok = (m < M);
    int bc = rowok ? blocks[m] : 0;

    v8f acc[4];
#pragma unroll
    for (int n = 0; n < 4; ++n)
#pragma unroll
        for (int e = 0; e < 8; ++e) acc[n][e] = 0.0f;

    for (int t = 0; t < 8; ++t) {
        int ri = rowok ? slots[bc * 8 + t] : -1;
        v2f a[8];
        if (ri >= 0) {
            const float* pA = feat + (long)ri * 32;
#pragma unroll
            for (int q = 0; q < 8; ++q) {
                int k = 4 * q + 2 * g;
                v2f av; av.x = pA[k]; av.y = pA[k + 1];
                a[q] = av;
            }
        } else {
#pragma unroll
            for (int q = 0; q < 8; ++q) { v2f av; av.x = 0.0f; av.y = 0.0f; a[q] = av; }
        }
        const float* pB = bpack + (size_t)t * 4 * 8 * 32 * 2;
#pragma unroll
        for (int n = 0; n < 4; ++n) {
#pragma unroll
            for (int q = 0; q < 8; ++q) {
                const float* pb = pB + ((n * 8 + q) * 32 + lane) * 2;
                v2f b; b.x = pb[0]; b.y = pb[1];
                acc[n] = __builtin_amdgcn_wmma_f32_16x16x4_f32(
                    false, a[q], false, b, (short)0, acc[n], false, false);
            }
        }
    }

    // D layout: VGPR rr -> M = m0 + rr + 8*g, N = n*16 + (lane&15)
#pragma unroll
    for (int n = 0; n < 4; ++n) {
#pragma unroll
        for (int rr = 0; rr < 8; ++rr) {
            int row = m0 + rr + 8 * g;
            if (row < M) outf[(size_t)row * 64 + n * 16 + (lane & 15)] = acc[n][rr];
        }
    }
}

// ---------------------------------------------------------------------------
extern "C" void kernel_launch(void* const* d_in, const int* in_sizes, int n_in,
                              void* d_out, int out_size, void* d_ws, size_t ws_size,
                              hipStream_t stream) {
    const float* feat = (const float*)d_in[0];   // [N,32] f32
    const int*   pos  = (const int*)d_in[1];     // [N,3]  i32
    const float* wgt  = (const float*)d_in[2];   // [2,2,2,32,64] f32

    int N = in_sizes[1] / 3;
    int M = out_size / 67;                       // M*64 feats + M*3 positions

    char* ws = (char*)d_ws;
    int*           slots    = (int*)(ws + 0);                                   // 8 MB
    int*           blocks   = (int*)(ws + (size_t)8 * 1024 * 1024);             // 1 MB
    unsigned char* occ      = (unsigned char*)(ws + (size_t)9 * 1024 * 1024);   // 256 KB
    int*           partials = (int*)(ws + (size_t)9 * 1024 * 1024 + 256 * 1024);// 1 KB
    float*         bpack    = (float*)(ws + (size_t)9 * 1024 * 1024 + 260 * 1024); // 64 KB

    float* outf = (float*)d_out;
    float* outp = outf + (size_t)M * 64;

    int nslots = NBLK * 8;
    k_init_slots<<<(nslots + 255) / 256, 256, 0, stream>>>(slots, nslots);
    k_scatter<<<(N + 255) / 256, 256, 0, stream>>>(pos, slots, N);
    k_count<<<NGRP, 256, 0, stream>>>(slots, occ, partials);
    k_compact<<<NGRP, 256, 0, stream>>>(occ, partials, blocks, outp, M);
    k_wpack<<<(8 * 4 * 8 * 32 + 255) / 256, 256, 0, stream>>>(wgt, bpack);

    int tiles = (M + 15) / 16;
    k_gemm<<<(tiles + 7) / 8, 256, 0, stream>>>(feat, slots, blocks, bpack, outf, M);
}